// RouteNgramMemory_24781961298265
// MI455X (gfx1250) — compile-verified
//
#include <hip/hip_runtime.h>
#include <hip/hip_bf16.h>

typedef __attribute__((ext_vector_type(2))) float v2f;
typedef __attribute__((ext_vector_type(8))) float v8f;

// Problem dims (fixed by the reference)
constexpr int Bb = 8, Tt = 4096, Rr = 8, NB = 4, Dd = 64;
constexpr int P    = Bb * Tt * Rr;   // 262144 positions
constexpr int TILE = 16;             // positions per wave (WMMA M dim)
constexpr int WAVES = 8;             // waves per block
constexpr int THREADS = WAVES * 32;
constexpr int LDS_STRIDE = 17;       // 16 + 1 pad -> conflict-free transpose

__global__ __launch_bounds__(THREADS)
void route_ngram_surrogate_kernel(const float* __restrict__ grad,      // [P,64]
                                  const float* __restrict__ plocal,    // [P,4]
                                  const float* __restrict__ table,     // [32768,64]
                                  const int*   __restrict__ addr_base, // [P]
                                  const int*   __restrict__ route_off, // [8]
                                  const int*   __restrict__ code_stride_p, // [1]
                                  float*       __restrict__ out)       // [P,4]
{
    __shared__ float lds[WAVES][TILE * LDS_STRIDE];

    const int lane = threadIdx.x & 31;
    const int wave = threadIdx.x >> 5;
    const int tile = blockIdx.x * WAVES + wave;   // 16 positions per tile
    const int c    = lane & 15;                   // code index (N / code lane)
    const int h    = lane >> 4;                   // D-half (0: dims 0..31, 1: 32..63)
    const int cs   = code_stride_p[0];            // 16

    float* wT = lds[wave];

    // ---------------- Phase 1: gathered dots + probabilities --------------
    // lanes = (code c, half h). Each lane reduces 32 dims; shfl_xor(16) folds
    // the two halves so lanes 0..15 hold dot_c exactly in f32.
    for (int m = 0; m < TILE; ++m) {
        const int p    = tile * TILE + m;
        const int r    = p & (Rr - 1);                       // layout [B,T,R]
        const int row  = route_off[r] + addr_base[p] + c * cs;
        const float4* e = (const float4*)(table + (size_t)row * Dd + h * 32);
        const float4* g = (const float4*)(grad  + (size_t)p   * Dd + h * 32);

        if (m + 1 < TILE) { // prefetch next grad row (global_prefetch_b8)
            __builtin_prefetch(grad + (size_t)(p + 1) * Dd + h * 32, 0, 1);
        }

        float acc = 0.f;
        #pragma unroll
        for (int k = 0; k < 8; ++k) {
            const float4 ev = e[k];
            const float4 gv = g[k];
            acc = fmaf(ev.x, gv.x, acc);
            acc = fmaf(ev.y, gv.y, acc);
            acc = fmaf(ev.z, gv.z, acc);
            acc = fmaf(ev.w, gv.w, acc);
        }
        const float dot = acc + __shfl_xor(acc, 16, 32);

        const float4 pl = *(const float4*)(plocal + (size_t)p * NB);
        float prob = ((c & 1) ? pl.x : 1.f - pl.x);
        prob      *= ((c & 2) ? pl.y : 1.f - pl.y);
        prob      *= ((c & 4) ? pl.z : 1.f - pl.z);
        prob      *= ((c & 8) ? pl.w : 1.f - pl.w);

        if (h == 0) wT[m * LDS_STRIDE + c] = prob * dot;     // W[m, c]
    }
    __syncthreads();

    // ---------------- Phase 2: exact f32 WMMA contraction -----------------
    // D[m, nb<4] = sum_c W[m,c]*bits[c,nb] ; D[m, 4] = sum_c W[m,c] (ones col)
    // K = 16 codes chunked into 4 x V_WMMA_F32_16X16X4_F32, accumulating C.
    v8f dacc = {};
    #pragma unroll
    for (int j = 0; j < 4; ++j) {
        const int k0 = 4 * j;
        // A (16x4 f32): lane m (0-15) holds K={0,1} in vgpr {0,1};
        //               lane m+16 holds K={2,3}.
        const int am = lane & 15;
        v2f a;
        a.x = wT[am * LDS_STRIDE + k0 + 2 * h + 0];
        a.y = wT[am * LDS_STRIDE + k0 + 2 * h + 1];
        // B (4x16 f32): vgpr0 = K rows {0|2} by half, vgpr1 = K rows {1|3};
        // N = lane&15. Columns: 0..3 = bits, 4 = ones (gives s0), rest 0.
        const int kx = k0 + 2 * h;
        const int n  = c;
        v2f b;
        b.x = (n < 4) ? (float)(((kx + 0) >> n) & 1) : (n == 4 ? 1.f : 0.f);
        b.y = (n < 4) ? (float)(((kx + 1) >> n) & 1) : (n == 4 ? 1.f : 0.f);

        dacc = __builtin_amdgcn_wmma_f32_16x16x4_f32(
            /*neg_a=*/false, a, /*neg_b=*/false, b,
            /*c_mod=*/(short)0, dacc, /*reuse_a=*/false, /*reuse_b=*/false);
    }
    __syncthreads();

    // Scatter D tile to LDS: vgpr v -> row M = v + 8*h, col N = lane&15.
    #pragma unroll
    for (int v = 0; v < 8; ++v) {
        wT[(v + 8 * h) * LDS_STRIDE + c] = dacc[v];
    }
    __syncthreads();

    // ---------------- Phase 3: epilogue + coalesced store -----------------
    // out[p, nb] = s1[p, nb] - p_local[p, nb] * s0[p]; 64 floats per tile.
    const size_t obase = (size_t)tile * (TILE * NB);
    #pragma unroll
    for (int q = 0; q < 2; ++q) {
        const int o  = lane + q * 32;
        const int m  = o >> 2;
        const int nb = o & 3;
        const float s1 = wT[m * LDS_STRIDE + nb];
        const float s0 = wT[m * LDS_STRIDE + 4];
        out[obase + o] = s1 - plocal[obase + o] * s0;
    }
}

extern "C" void kernel_launch(void* const* d_in, const int* in_sizes, int n_in,
                              void* d_out, int out_size, void* d_ws, size_t ws_size,
                              hipStream_t stream) {
    (void)in_sizes; (void)n_in; (void)out_size; (void)d_ws; (void)ws_size;
    const float* grad       = (const float*)d_in[0];  // grad_valid [B,T,R,D]
    const float* plocal     = (const float*)d_in[1];  // p_local    [B,T,R,NB]
    const float* table      = (const float*)d_in[2];  // table_weight [32768,64]
    // d_in[3] = code_bits [16,4] -- recomputed from lane bits, unused
    const int*   addr_base  = (const int*)d_in[4];    // [B,T,R]
    const int*   route_off  = (const int*)d_in[5];    // [8]
    const int*   code_str   = (const int*)d_in[6];    // scalar 16
    float*       out        = (float*)d_out;          // [B,T,R,NB]

    const int tiles  = P / TILE;          // 16384
    const int blocks = tiles / WAVES;     // 2048
    route_ngram_surrogate_kernel<<<blocks, THREADS, 0, stream>>>(
        grad, plocal, table, addr_base, route_off, code_str, out);
}